// NeuralMemory_51831665328611
// MI455X (gfx1250) — compile-verified
//
#include <hip/hip_runtime.h>

#define N_SLOTS 65536
#define SLOT_D  256
#define BATCH   512
#define KTOP    64
#define NKCHUNK 16
#define KB_STAGE 64   // K rows of B staged in LDS per step (64*256*4B = 64KB)

constexpr float BETA_CONTENT = 1.0f;
constexpr float BETA_TOPK    = 10.0f;
constexpr float ERASE_F      = 0.5f;
constexpr float EPS_F        = 1e-8f;
constexpr float NEG_BIG      = -1000000000.0f;

typedef __attribute__((ext_vector_type(2))) float v2f;
typedef __attribute__((ext_vector_type(4))) float v4f;
typedef __attribute__((ext_vector_type(8))) float v8f;

// ---------------- WMMA wrapper: D = A(16x4,f32) * B(4x16,f32) + C ----------------
__device__ __forceinline__ v8f wmma_f32(v2f a, v2f b, v8f c) {
    return __builtin_amdgcn_wmma_f32_16x16x4_f32(
        /*neg_a=*/false, a, /*neg_b=*/false, b,
        /*c_mod=*/(short)0, c, /*reuse_a=*/false, /*reuse_b=*/false);
}

// ---------------- CDNA5 async global->LDS copy (ASYNCcnt-tracked) ----------------
__device__ __forceinline__ void async_copy_b128(float* ldsDst, const float* gsrc) {
    unsigned int ldsOff = (unsigned int)(unsigned long long)ldsDst; // addrspace(3) offset
    asm volatile("global_load_async_to_lds_b128 %0, %1, off"
                 :: "v"(ldsOff), "v"(gsrc) : "memory");
}
__device__ __forceinline__ void wait_asynccnt0() {
    asm volatile("s_wait_asynccnt 0" ::: "memory");
}

// ---------------- wave32 / block reductions ----------------
__device__ __forceinline__ float waveSum(float v) {
    #pragma unroll
    for (int o = 16; o; o >>= 1) v += __shfl_xor(v, o, 32);
    return v;
}
__device__ __forceinline__ float waveMax(float v) {
    #pragma unroll
    for (int o = 16; o; o >>= 1) v = fmaxf(v, __shfl_xor(v, o, 32));
    return v;
}
__device__ __forceinline__ int waveSumI(int v) {
    #pragma unroll
    for (int o = 16; o; o >>= 1) v += __shfl_xor(v, o, 32);
    return v;
}

// blockDim.x must be a multiple of 32, <= 1024 (<= 32 waves)
__device__ __forceinline__ float blockSum(float v, float* sm) {
    int lane = threadIdx.x & 31, w = threadIdx.x >> 5, nw = blockDim.x >> 5;
    v = waveSum(v);
    if (lane == 0) sm[w] = v;
    __syncthreads();
    if (w == 0) {
        float t = (lane < nw) ? sm[lane] : 0.0f;
        t = waveSum(t);
        if (lane == 0) sm[0] = t;
    }
    __syncthreads();
    float r = sm[0];
    __syncthreads();
    return r;
}
__device__ __forceinline__ float blockMax(float v, float* sm) {
    int lane = threadIdx.x & 31, w = threadIdx.x >> 5, nw = blockDim.x >> 5;
    v = waveMax(v);
    if (lane == 0) sm[w] = v;
    __syncthreads();
    if (w == 0) {
        float t = (lane < nw) ? sm[lane] : -3.4e38f;
        t = waveMax(t);
        if (lane == 0) sm[0] = t;
    }
    __syncthreads();
    float r = sm[0];
    __syncthreads();
    return r;
}
__device__ __forceinline__ int blockSumI(int v, int* sm) {
    int lane = threadIdx.x & 31, w = threadIdx.x >> 5, nw = blockDim.x >> 5;
    v = waveSumI(v);
    if (lane == 0) sm[w] = v;
    __syncthreads();
    if (w == 0) {
        int t = (lane < nw) ? sm[lane] : 0;
        t = waveSumI(t);
        if (lane == 0) sm[0] = t;
    }
    __syncthreads();
    int r = sm[0];
    __syncthreads();
    return r;
}

// ---------------- K1: L2-normalize rows (one wave per row of 256) ----------------
__global__ void norm_rows_kernel(const float* __restrict__ src, float* __restrict__ dst, int rows) {
    int gwave = (blockIdx.x * blockDim.x + threadIdx.x) >> 5;
    int lane  = threadIdx.x & 31;
    if (gwave >= rows) return;
    const float* r = src + (long)gwave * SLOT_D;
    float v[8]; float ss = 0.0f;
    #pragma unroll
    for (int j = 0; j < 8; ++j) { v[j] = r[lane + 32 * j]; ss += v[j] * v[j]; }
    ss = waveSum(ss);
    float inv = 1.0f / fmaxf(sqrtf(ss), 1e-12f);
    float* o = dst + (long)gwave * SLOT_D;
    #pragma unroll
    for (int j = 0; j < 8; ++j) o[lane + 32 * j] = v[j] * inv;
}

// ---------------- K2: sim = qn(512x256) @ mn^T(256x65536) via WMMA f32 16x16x4 ----
// grid = (4096/8 Ntile-groups, 32 Mtiles), block = 256 (8 waves, 1 tile/wave)
__global__ void sim_gemm_kernel(const float* __restrict__ qn, const float* __restrict__ mn,
                                float* __restrict__ sim) {
    int wave = threadIdx.x >> 5, lane = threadIdx.x & 31;
    int lr = lane & 15, hi = lane >> 4;
    int tileM = blockIdx.y * 16;
    int tileN = (blockIdx.x * 8 + wave) * 16;
    v8f c = {};
    const float* aRow = qn + (long)(tileM + lr) * SLOT_D;   // A[m][k] = qn[m*256+k]
    const float* bRow = mn + (long)(tileN + lr) * SLOT_D;   // B[k][n] = mn[n*256+k]
    bool sel = (hi != 0);
    for (int k0 = 0; k0 < SLOT_D; k0 += 4) {
        v2f a = *(const v2f*)(aRow + k0 + 2 * hi);          // K pair {k0+2hi, k0+2hi+1}
        v4f bc = *(const v4f*)(bRow + k0);                  // one b128 instead of 2x b32
        v2f b;
        b.x = sel ? bc.y : bc.x;                            // K = k0+hi
        b.y = sel ? bc.w : bc.z;                            // K = k0+2+hi
        c = wmma_f32(a, b, c);
    }
    #pragma unroll
    for (int r = 0; r < 8; ++r)
        sim[(long)(tileM + r + 8 * hi) * N_SLOTS + tileN + lr] = c[r];
}

// ---------------- K3: per-row softmax + top-k-threshold softmax --------------------
// grid = 512 rows, block = 1024 (each thread owns 64 strided elements in registers)
__global__ __launch_bounds__(1024) void softmax_topk_kernel(float* __restrict__ sim,      // in: sim, out: w_content
                                                            float* __restrict__ w_topk) {
    __shared__ float smf[32];
    __shared__ int   smi[32];
    int row = blockIdx.x, tid = threadIdx.x;
    float* srow = sim    + (long)row * N_SLOTS;
    float* trow = w_topk + (long)row * N_SLOTS;

    float vals[64]; float vmax = -3.4e38f;
    #pragma unroll
    for (int j = 0; j < 64; ++j) { vals[j] = srow[tid + j * 1024]; vmax = fmaxf(vmax, vals[j]); }
    vmax = blockMax(vmax, smf);

    // bisection for the K-th largest value: thr = largest t with count(x>=t) >= KTOP
    float lo = -2.0f, hi = 2.0f;
    for (int it = 0; it < 40; ++it) {
        float mid = 0.5f * (lo + hi);
        int cnt = 0;
        #pragma unroll
        for (int j = 0; j < 64; ++j) cnt += (vals[j] >= mid) ? 1 : 0;
        cnt = blockSumI(cnt, smi);
        if (cnt >= KTOP) lo = mid; else hi = mid;
    }
    float thr = lo;

    // content softmax (beta = 1), normalized in place over sim buffer
    float cmax = BETA_CONTENT * vmax;
    float sc = 0.0f;
    #pragma unroll
    for (int j = 0; j < 64; ++j) sc += __expf(BETA_CONTENT * vals[j] - cmax);
    sc = blockSum(sc, smf);
    float invc = 1.0f / sc;
    #pragma unroll
    for (int j = 0; j < 64; ++j) srow[tid + j * 1024] = __expf(BETA_CONTENT * vals[j] - cmax) * invc;

    // top-k gated softmax: t = beta*sim + (1 - sigmoid(beta*(sim-thr)))*NEG
    float tmax = -3.4e38f;
    #pragma unroll
    for (int j = 0; j < 64; ++j) {
        float d = BETA_TOPK * (vals[j] - thr);
        float mask = 1.0f / (1.0f + __expf(-d));
        float t = BETA_TOPK * vals[j] + (1.0f - mask) * NEG_BIG;
        vals[j] = t;
        tmax = fmaxf(tmax, t);
    }
    tmax = blockMax(tmax, smf);
    float st = 0.0f;
    #pragma unroll
    for (int j = 0; j < 64; ++j) { float e = __expf(vals[j] - tmax); vals[j] = e; st += e; }
    st = blockSum(st, smf);
    float invt = 1.0f / st;
    #pragma unroll
    for (int j = 0; j < 64; ++j) trow[tid + j * 1024] = vals[j] * invt;
}

// ---------------- K4a: partial read GEMMs (both w_content & w_topk vs memory) ------
// grid = (NKCHUNK, 32 Mtiles), block = 256; B (memory K-chunk) staged in LDS via
// CDNA5 async global->LDS copies; wave handles 2 N-tiles, 4 WMMAs/K-step
__global__ __launch_bounds__(256) void read_gemm_partial_kernel(
        const float* __restrict__ wc, const float* __restrict__ wt,
        const float* __restrict__ mem, float* __restrict__ part) {
    __shared__ float ldsB[KB_STAGE * SLOT_D];   // 64KB
    int wave = threadIdx.x >> 5, lane = threadIdx.x & 31;
    int lr = lane & 15, hi = lane >> 4;
    int tileM = blockIdx.y * 16;
    int kBeg = blockIdx.x * (N_SLOTS / NKCHUNK);
    int kEnd = kBeg + (N_SLOTS / NKCHUNK);
    int n0 = wave * 32;
    v8f c0 = {}, c1 = {}, t0 = {}, t1 = {};
    for (int kc = kBeg; kc < kEnd; kc += KB_STAGE) {
        // cooperative async stage of B chunk: KB_STAGE x 256 f32 (4096 x b128)
        #pragma unroll
        for (int it = 0; it < (KB_STAGE * SLOT_D) / (256 * 4); ++it) {
            int idx = (it * 256 + threadIdx.x) * 4;   // float index, 16B granules
            async_copy_b128(&ldsB[idx], mem + (long)kc * SLOT_D + idx);
        }
        wait_asynccnt0();
        __syncthreads();
        for (int kk = 0; kk < KB_STAGE; kk += 4) {
            long arow = (long)(tileM + lr) * N_SLOTS + kc + kk + 2 * hi;
            v2f ac = *(const v2f*)(wc + arow);
            v2f at = *(const v2f*)(wt + arow);
            const float* bp = &ldsB[(kk + hi) * SLOT_D];
            v2f b0, b1;
            b0.x = bp[n0 + lr];                b1.x = bp[n0 + 16 + lr];
            b0.y = bp[2 * SLOT_D + n0 + lr];   b1.y = bp[2 * SLOT_D + n0 + 16 + lr];
            c0 = wmma_f32(ac, b0, c0);  c1 = wmma_f32(ac, b1, c1);
            t0 = wmma_f32(at, b0, t0);  t1 = wmma_f32(at, b1, t1);
        }
        __syncthreads();
    }
    float* pc = part + (long)blockIdx.x * (BATCH * SLOT_D);
    float* pt = part + (long)(NKCHUNK + blockIdx.x) * (BATCH * SLOT_D);
    #pragma unroll
    for (int r = 0; r < 8; ++r) {
        long m = tileM + r + 8 * hi;
        pc[m * SLOT_D + n0 + lr]      = c0[r];
        pc[m * SLOT_D + n0 + 16 + lr] = c1[r];
        pt[m * SLOT_D + n0 + lr]      = t0[r];
        pt[m * SLOT_D + n0 + 16 + lr] = t1[r];
    }
}

// ---------------- K4b: deterministic partial reduction -> d_out[0 .. 262144) -------
__global__ void read_reduce_kernel(const float* __restrict__ part, float* __restrict__ out) {
    long i = (long)blockIdx.x * blockDim.x + threadIdx.x;
    if (i >= 2L * BATCH * SLOT_D) return;
    int which = (i >= (long)BATCH * SLOT_D) ? 1 : 0;
    long e = i - (long)which * BATCH * SLOT_D;
    float s = 0.0f;
    #pragma unroll
    for (int k = 0; k < NKCHUNK; ++k)
        s += part[((long)which * NKCHUNK + k) * (BATCH * SLOT_D) + e];
    out[i] = s;
}

// ---------------- K5: shift -> sharpen -> normalize -> softmax => w_write ----------
// grid = 512 rows, block = 1024
__global__ __launch_bounds__(1024) void shift_sharpen_kernel(const float* __restrict__ prev,
                                                             const float* __restrict__ sw,
                                                             float* __restrict__ w_write) {
    __shared__ float smf[32];
    int row = blockIdx.x, tid = threadIdx.x;
    float s0 = sw[0], s1 = sw[1], s2 = sw[2];
    const float* p = prev + (long)row * N_SLOTS;
    float* o = w_write + (long)row * N_SLOTS;
    float sh[64]; float sum = 0.0f, mx = -3.4e38f;
    #pragma unroll
    for (int j = 0; j < 64; ++j) {
        int i = tid + j * 1024;
        float pm1 = (i > 0) ? p[i - 1] : 0.0f;
        float pc  = p[i];
        float pp1 = (i < N_SLOTS - 1) ? p[i + 1] : 0.0f;
        float s = fmaxf(pm1 * s0 + pc * s1 + pp1 * s2, 0.0f);
        float v = s * sqrtf(s);            // s^1.5 (GAMMA = 1.5)
        sh[j] = v; sum += v; mx = fmaxf(mx, v);
    }
    sum = blockSum(sum, smf);
    mx  = blockMax(mx, smf);
    float inv = 1.0f / (sum + EPS_F);
    float wmax = mx * inv;
    float se = 0.0f;
    #pragma unroll
    for (int j = 0; j < 64; ++j) { float e = __expf(sh[j] * inv - wmax); sh[j] = e; se += e; }
    se = blockSum(se, smf);
    float rn = 1.0f / se;
    #pragma unroll
    for (int j = 0; j < 64; ++j) o[tid + j * 1024] = sh[j] * rn;
}

// ---------------- K6: erase_factor[n] = prod_b (1 - w_write[b][n]*ERASE) -----------
__global__ void erase_kernel(const float* __restrict__ ww, float* __restrict__ er) {
    int n = blockIdx.x * blockDim.x + threadIdx.x;
    if (n >= N_SLOTS) return;
    float pr = 1.0f;
    for (int b = 0; b < BATCH; ++b)
        pr *= (1.0f - ww[(long)b * N_SLOTS + n] * ERASE_F);
    er[n] = pr;
}

// ---------------- K7: new_mem = memory*erase + w_write^T @ value -------------------
// grid = 4096 Mtiles, block = 256; wave handles 2 N-tiles; K = 512 (128 steps)
__global__ void write_gemm_kernel(const float* __restrict__ ww, const float* __restrict__ val,
                                  const float* __restrict__ mem, const float* __restrict__ er,
                                  float* __restrict__ out) {
    int wave = threadIdx.x >> 5, lane = threadIdx.x & 31;
    int lr = lane & 15, hi = lane >> 4;
    int tileM = blockIdx.x * 16;
    int n0 = wave * 32;
    v8f c0 = {}, c1 = {};
    for (int k0 = 0; k0 < BATCH; k0 += 4) {
        v2f a;  // A[m][k] = ww[k*N_SLOTS + m]
        a.x = ww[(long)(k0 + 2 * hi)     * N_SLOTS + tileM + lr];
        a.y = ww[(long)(k0 + 2 * hi + 1) * N_SLOTS + tileM + lr];
        const float* b0p = val + (long)(k0 + hi) * SLOT_D;
        v2f b0, b1;
        b0.x = b0p[n0 + lr];                b1.x = b0p[n0 + 16 + lr];
        b0.y = b0p[2 * SLOT_D + n0 + lr];   b1.y = b0p[2 * SLOT_D + n0 + 16 + lr];
        c0 = wmma_f32(a, b0, c0);
        c1 = wmma_f32(a, b1, c1);
    }
    #pragma unroll
    for (int r = 0; r < 8; ++r) {
        long m = tileM + r + 8 * hi;
        float ef = er[m];
        out[m * SLOT_D + n0 + lr]      = mem[m * SLOT_D + n0 + lr]      * ef + c0[r];
        out[m * SLOT_D + n0 + 16 + lr] = mem[m * SLOT_D + n0 + 16 + lr] * ef + c1[r];
    }
}

extern "C" void kernel_launch(void* const* d_in, const int* in_sizes, int n_in,
                              void* d_out, int out_size, void* d_ws, size_t ws_size,
                              hipStream_t stream) {
    const float* memory = (const float*)d_in[0];   // 65536 x 256
    const float* query  = (const float*)d_in[1];   // 512 x 256
    const float* value  = (const float*)d_in[2];   // 512 x 256
    const float* prevw  = (const float*)d_in[3];   // 512 x 65536
    const float* shiftw = (const float*)d_in[4];   // 3
    // d_in[5] = k (compile-time KTOP = 64)

    float* ws = (float*)d_ws;
    float* qn      = ws;                                   // 131072
    float* mn      = qn + (long)BATCH * SLOT_D;            // 16777216
    float* simbuf  = mn + (long)N_SLOTS * SLOT_D;          // 33554432 (sim -> w_content -> reused as w_write)
    float* wtopk   = simbuf + (long)BATCH * N_SLOTS;       // 33554432
    float* part    = wtopk + (long)BATCH * N_SLOTS;        // 2*NKCHUNK*131072 = 4194304
    float* erasef  = part + 2L * NKCHUNK * BATCH * SLOT_D; // 65536
    float* wwrite  = simbuf;                               // safe: reused after read GEMM (stream order)

    float* out_reads = (float*)d_out;                      // [read_content | read_topk] = 262144
    float* out_mem   = out_reads + 2L * BATCH * SLOT_D;    // new_mem = 16777216

    // K1: normalize
    norm_rows_kernel<<<N_SLOTS / 8, 256, 0, stream>>>(memory, mn, N_SLOTS);
    norm_rows_kernel<<<BATCH / 8,  256, 0, stream>>>(query,  qn, BATCH);

    // K2: sim GEMM
    sim_gemm_kernel<<<dim3(N_SLOTS / 16 / 8, BATCH / 16), 256, 0, stream>>>(qn, mn, simbuf);

    // K3: row softmax + top-k softmax (sim -> w_content in place, w_topk out)
    softmax_topk_kernel<<<BATCH, 1024, 0, stream>>>(simbuf, wtopk);

    // K4: read GEMMs (partials over K chunks, then deterministic reduce)
    read_gemm_partial_kernel<<<dim3(NKCHUNK, BATCH / 16), 256, 0, stream>>>(simbuf, wtopk, memory, part);
    read_reduce_kernel<<<(2 * BATCH * SLOT_D + 255) / 256, 256, 0, stream>>>(part, out_reads);

    // K5: shift/sharpen/softmax -> w_write (reuses sim buffer region)
    shift_sharpen_kernel<<<BATCH, 1024, 0, stream>>>(prevw, shiftw, wwrite);

    // K6: erase factors
    erase_kernel<<<N_SLOTS / 256, 256, 0, stream>>>(wwrite, erasef);

    // K7: new_mem = memory*erase + w_write^T @ value
    write_gemm_kernel<<<N_SLOTS / 16, 256, 0, stream>>>(wwrite, value, memory, erasef, out_mem);
}